// Fourier_DiT_3D_9552007266336
// MI455X (gfx1250) — compile-verified
//
#include <hip/hip_runtime.h>
#include <hip/hip_bf16.h>

// Fourier-DiT-3D for MI455X (gfx1250, wave32).
// Roofline: r0..r2 stream 3x128MB = 384MB -> ~16.5us floor @ 23.3TB/s.
// Spectral einsum is 16 flop/byte @ fp32 -> V_WMMA_F32_16X16X4_F32 everywhere
// (data is complex64; the streamed operand is f32 so low-precision buys nothing).
// FFTs are DFT-matrix WMMA GEMMs staged in LDS; pointwise 64x64 complex linears
// are WMMA GEMMs fused with exp(z - shrink) activations.

#define NB   16
#define FDIM 4096                 // T*H*W
#define PTS  (NB * FDIM)          // 65536 points
#define HIDD 64

typedef __attribute__((ext_vector_type(2))) float v2f;
typedef __attribute__((ext_vector_type(8))) float v8f;

__device__ __forceinline__ v8f wmma_f32(v2f a, v2f b, v8f c) {
  // D[16x16] = A[16x4] * B[4x16] + C   (f32, wave32)
  return __builtin_amdgcn_wmma_f32_16x16x4_f32(false, a, false, b, (short)0, c, false, false);
}

// complex accumulate: (dre + i*dim) += (are + i*aim) * (bre + i*bim)
__device__ __forceinline__ void cwmma(v8f& dre, v8f& dim, v2f are, v2f aim, v2f bre, v2f bim) {
  dre = wmma_f32(are, bre, dre);
  v2f nbim; nbim.x = -bim.x; nbim.y = -bim.y;   // f32 WMMA has no A/B NEG bits
  dre = wmma_f32(aim, nbim, dre);
  dim = wmma_f32(are, bim, dim);
  dim = wmma_f32(aim, bre, dim);
}

__device__ __forceinline__ void euler_act(float& re, float& im, float shrink) {
  const float e = __expf(re - shrink);
  float s, c; __sincosf(im, &s, &c);
  re = e * c; im = e * s;
}

// Wave-level complex GEMM: D[16 x 64] += A[16 x 64] @ W^T, A from LDS (re/im
// split, row-major [m][i]), W is [O][64] complex row-major (torch Linear).
__device__ __forceinline__ void cgemm64(const float* Are, const float* Aim,
                                        const float2* W, int obase,
                                        v8f cre[4], v8f cim[4],
                                        int l15, int lh) {
  for (int k0 = 0; k0 < HIDD; k0 += 4) {
    const int i = k0 + 2 * lh;
    v2f are; are.x = Are[l15 * HIDD + i]; are.y = Are[l15 * HIDD + i + 1];
    v2f aim; aim.x = Aim[l15 * HIDD + i]; aim.y = Aim[l15 * HIDD + i + 1];
#pragma unroll
    for (int ot = 0; ot < 4; ot++) {
      const int o = obase + ot * 16 + l15;
      const float2 w0 = W[(size_t)o * HIDD + i];
      const float2 w1 = W[(size_t)o * HIDD + i + 1];
      v2f bre; bre.x = w0.x; bre.y = w1.x;
      v2f bim; bim.x = w0.y; bim.y = w1.y;
      cwmma(cre[ot], cim[ot], are, aim, bre, bim);
    }
  }
}

// ---------------------------------------------------------------------------
// Time-modulation branch: tp = lin(act(lin(act(i*t*t_w), tm0)), tm1) -> 3 chunks
// ---------------------------------------------------------------------------
__global__ void k_tp(const float* __restrict__ t, const float2* __restrict__ t_w,
                     const float2* __restrict__ tm0_w, const float2* __restrict__ tm0_b,
                     const float2* __restrict__ tm1_w, const float2* __restrict__ tm1_b,
                     float2* __restrict__ tp) {
  __shared__ float lds[2][2][16][HIDD];
  const int lane = threadIdx.x & 31, wv = threadIdx.x >> 5;
  const int l15 = lane & 15, lh = lane >> 4;
  const int p0 = (blockIdx.x * 2 + wv) * 16;
  float* Lre = &lds[wv][0][0][0];
  float* Lim = &lds[wv][1][0][0];

  const int pA = p0 + l15;
  const int nA = pA >> 12;
  const int fA = pA & (FDIM - 1);
  const float tn = t[nA];
  const float E1 = 0.3678794411714423f; // e^-1

  const v8f vz = {0.f,0.f,0.f,0.f,0.f,0.f,0.f,0.f};
  v8f cre[4], cim[4];
  for (int i = 0; i < 4; i++) { cre[i] = vz; cim[i] = vz; }

  // GEMM1: act(i*t*t_w_im) @ tm0_w^T    (A fragments generated analytically)
  for (int k0 = 0; k0 < HIDD; k0 += 4) {
    const int i = k0 + 2 * lh;
    const float th0 = tn * t_w[(size_t)fA * HIDD + i].y;      // t_w purely imaginary
    const float th1 = tn * t_w[(size_t)fA * HIDD + i + 1].y;
    float s0, c0, s1, c1;
    __sincosf(th0, &s0, &c0); __sincosf(th1, &s1, &c1);
    v2f are; are.x = E1 * c0; are.y = E1 * c1;
    v2f aim; aim.x = E1 * s0; aim.y = E1 * s1;
#pragma unroll
    for (int ot = 0; ot < 4; ot++) {
      const int o = ot * 16 + l15;
      const float2 w0 = tm0_w[(size_t)o * HIDD + i];
      const float2 w1 = tm0_w[(size_t)o * HIDD + i + 1];
      v2f bre; bre.x = w0.x; bre.y = w1.x;
      v2f bim; bim.x = w0.y; bim.y = w1.y;
      cwmma(cre[ot], cim[ot], are, aim, bre, bim);
    }
  }
  // bias + act, stage to LDS in A-layout for GEMM2
#pragma unroll
  for (int ot = 0; ot < 4; ot++) {
    const int o = ot * 16 + l15;
    const float2 b = tm0_b[o];
#pragma unroll
    for (int v = 0; v < 8; v++) {
      const int m = v + 8 * lh;
      float re = cre[ot][v] + b.x, im = cim[ot][v] + b.y;
      euler_act(re, im, 1.0f);
      Lre[m * HIDD + o] = re; Lim[m * HIDD + o] = im;
    }
  }
  __syncthreads();
  // GEMM2: 192 outputs in 3 passes of 64
  for (int j = 0; j < 3; j++) {
    for (int i = 0; i < 4; i++) { cre[i] = vz; cim[i] = vz; }
    cgemm64(Lre, Lim, tm1_w, j * HIDD, cre, cim, l15, lh);
#pragma unroll
    for (int ot = 0; ot < 4; ot++) {
      const int og = j * HIDD + ot * 16 + l15;
      const float2 b = tm1_b[og];
#pragma unroll
      for (int v = 0; v < 8; v++) {
        const int m = v + 8 * lh;
        const size_t p = (size_t)(p0 + m);
        tp[(size_t)j * PTS * HIDD + p * HIDD + (ot * 16 + l15)] =
            make_float2(cre[ot][v] + b.x, cim[ot][v] + b.y);
      }
    }
  }
}

// ---------------------------------------------------------------------------
// Input projection (K=3, VALU): xc[p][o] = sum_d x[p][d]*proj_w[o][d] + b[o]
// Writes point-major xc and channel-major xcT (FFT input layout).
// ---------------------------------------------------------------------------
__global__ void k_proj(const float* __restrict__ x,
                       const float2* __restrict__ proj_w, const float2* __restrict__ proj_b,
                       float2* __restrict__ xc, float2* __restrict__ xcT) {
  const int p = blockIdx.x * blockDim.x + threadIdx.x;
  if (p >= PTS) return;
  const int n = p >> 12, f = p & (FDIM - 1);
  const int tt = f >> 8, hw = f & 255;
  float xv[3];
#pragma unroll
  for (int d = 0; d < 3; d++) xv[d] = x[(((size_t)n * 16 + tt) * 3 + d) * 256 + hw];
  for (int o = 0; o < HIDD; o++) {
    const float2 b = proj_b[o];
    float re = b.x, im = b.y;
#pragma unroll
    for (int d = 0; d < 3; d++) {
      const float2 w = proj_w[o * 3 + d];
      re += xv[d] * w.x; im += xv[d] * w.y;
    }
    xc[(size_t)p * HIDD + o] = make_float2(re, im);
    xcT[((size_t)n * HIDD + o) * FDIM + f] = make_float2(re, im);
  }
}

// ---------------------------------------------------------------------------
// Fused 3D FFT (16^3) per (n,channel): three DFT-matrix WMMA passes in LDS.
// MODE 0: forward, out = zf[f][D][n].  MODE 1: inverse (conj, 1/4096),
//          out = spec point-major [n*F + f][k].
// ---------------------------------------------------------------------------
__device__ __forceinline__ int fft_addr(int pass, int line, int k) {
  if (pass == 0) return line * 16 + k;                            // W axis
  if (pass == 1) return (line >> 4) * 256 + k * 16 + (line & 15); // H axis
  return k * 256 + line;                                          // T axis
}

template <int MODE>
__global__ void k_fft(const float2* __restrict__ in, float2* __restrict__ out) {
  __shared__ float re[FDIM];
  __shared__ float im[FDIM];
  const int chan = blockIdx.x;                // n*64 + d
  const int n = chan >> 6, d = chan & 63;
  const float2* src = in + (size_t)chan * FDIM;
  for (int i = threadIdx.x; i < FDIM; i += blockDim.x) {
    const float2 v = src[i];
    re[i] = v.x; im[i] = v.y;
  }
  __syncthreads();
  const int lane = threadIdx.x & 31, wv = threadIdx.x >> 5;   // 8 waves
  const int l15 = lane & 15, lh = lane >> 4;
  const float SGN = MODE ? 1.0f : -1.0f;
  const float AMP = MODE ? 0.0625f : 1.0f;    // (1/16)^3 = ifft 1/4096
  // DFT matrix fragments: F[m][k] = AMP * exp(SGN * i*2pi*m*k/16), m = l15
  v2f fre[4], fim[4];
#pragma unroll
  for (int s = 0; s < 4; s++) {
    const int k = 4 * s + 2 * lh;
    const float w = SGN * 0.39269908169872414f * (float)l15;   // +-2pi/16 * m
    float s0, c0, s1, c1;
    __sincosf(w * (float)k, &s0, &c0);
    __sincosf(w * (float)(k + 1), &s1, &c1);
    fre[s].x = AMP * c0; fre[s].y = AMP * c1;
    fim[s].x = AMP * s0; fim[s].y = AMP * s1;
  }
  const v8f vz = {0.f,0.f,0.f,0.f,0.f,0.f,0.f,0.f};
  for (int pass = 0; pass < 3; pass++) {
    for (int lt = wv; lt < 16; lt += 8) {     // 16 line-tiles, 8 waves
      v8f cr = vz, ci = vz;
#pragma unroll
      for (int s = 0; s < 4; s++) {
        const int k = 4 * s + 2 * lh;
        const int line = lt * 16 + l15;
        const int a0 = fft_addr(pass, line, k);
        const int a1 = fft_addr(pass, line, k + 1);
        v2f bre; bre.x = re[a0]; bre.y = re[a1];
        v2f bim; bim.x = im[a0]; bim.y = im[a1];
        cwmma(cr, ci, fre[s], fim[s], bre, bim);
      }
      // in-place store (reads for this tile already done; tiles disjoint)
#pragma unroll
      for (int v = 0; v < 8; v++) {
        const int m = v + 8 * lh;
        const int line = lt * 16 + l15;
        const int a = fft_addr(pass, line, m);
        re[a] = cr[v]; im[a] = ci[v];
      }
    }
    __syncthreads();
  }
  if (MODE == 0) {
    for (int i = threadIdx.x; i < FDIM; i += blockDim.x)
      out[((size_t)i * HIDD + d) * NB + n] = make_float2(re[i], im[i]);
  } else {
    for (int i = threadIdx.x; i < FDIM; i += blockDim.x)
      out[((size_t)n * FDIM + i) * HIDD + d] = make_float2(re[i], im[i]);
  }
}

// ---------------------------------------------------------------------------
// Spectral mix (the 128MB/layer stream): per bin f,
//   zr[n,k,f] = (1/64) * sum_D zf[n,D,f] * r[D,f,k]
// One wave per bin, 16x64x64 complex GEMM, r read as coalesced 128B rows.
// ---------------------------------------------------------------------------
__global__ void k_spectral(const float2* __restrict__ zf,  // [f][64][16]
                           const float2* __restrict__ r,   // [64][4096][64]
                           float2* __restrict__ zr) {      // [n][k][4096]
  const int lane = threadIdx.x & 31, wv = threadIdx.x >> 5;
  const int l15 = lane & 15, lh = lane >> 4;
  const int f = blockIdx.x * 2 + wv;
  const v8f vz = {0.f,0.f,0.f,0.f,0.f,0.f,0.f,0.f};
  v8f cre[4], cim[4];
  for (int i = 0; i < 4; i++) { cre[i] = vz; cim[i] = vz; }
  const float2* zfb = zf + (size_t)f * HIDD * NB;
  for (int k0 = 0; k0 < HIDD; k0 += 4) {
    const int d = k0 + 2 * lh;
    const float2 a0 = zfb[d * NB + l15];
    const float2 a1 = zfb[(d + 1) * NB + l15];
    v2f are; are.x = a0.x; are.y = a1.x;
    v2f aim; aim.x = a0.y; aim.y = a1.y;
    const float2* rb0 = r + ((size_t)d * FDIM + f) * HIDD;
    const float2* rb1 = rb0 + (size_t)FDIM * HIDD;
    if (k0 + 4 < HIDD)                                        // stream-ahead
      __builtin_prefetch(rb0 + 4 * (size_t)FDIM * HIDD + lane, 0, 1);
#pragma unroll
    for (int ot = 0; ot < 4; ot++) {
      const float2 b0 = rb0[ot * 16 + l15];
      const float2 b1 = rb1[ot * 16 + l15];
      v2f bre; bre.x = b0.x; bre.y = b1.x;
      v2f bim; bim.x = b0.y; bim.y = b1.y;
      cwmma(cre[ot], cim[ot], are, aim, bre, bim);
    }
  }
  const float SC = 1.0f / 64.0f;
#pragma unroll
  for (int ot = 0; ot < 4; ot++)
#pragma unroll
    for (int v = 0; v < 8; v++) {
      const int nrow = v + 8 * lh, kcol = ot * 16 + l15;
      zr[((size_t)nrow * HIDD + kcol) * FDIM + f] =
          make_float2(SC * cre[ot][v], SC * cim[ot][v]);
    }
}

// ---------------------------------------------------------------------------
// Layer pointwise: xc' = exp(spec + xc@wr^T + wr_b + tp_i - 5); writes both
// point-major xc' and channel-major xcT for the next FFT.
// ---------------------------------------------------------------------------
__global__ void k_layeract(const float2* __restrict__ xc,
                           const float2* __restrict__ wr_w, const float2* __restrict__ wr_b,
                           const float2* __restrict__ spec, const float2* __restrict__ tpi,
                           float2* __restrict__ xco, float2* __restrict__ xcT) {
  __shared__ float lds[2][2][16][HIDD];
  const int lane = threadIdx.x & 31, wv = threadIdx.x >> 5;
  const int l15 = lane & 15, lh = lane >> 4;
  const int p0 = (blockIdx.x * 2 + wv) * 16;
  float* Lre = &lds[wv][0][0][0];
  float* Lim = &lds[wv][1][0][0];
  for (int e = lane; e < 16 * HIDD; e += 32) {
    const float2 v = xc[(size_t)p0 * HIDD + e];
    Lre[e] = v.x; Lim[e] = v.y;
  }
  __syncthreads();
  const v8f vz = {0.f,0.f,0.f,0.f,0.f,0.f,0.f,0.f};
  v8f cre[4], cim[4];
  for (int i = 0; i < 4; i++) { cre[i] = vz; cim[i] = vz; }
  cgemm64(Lre, Lim, wr_w, 0, cre, cim, l15, lh);
#pragma unroll
  for (int ot = 0; ot < 4; ot++) {
    const int o = ot * 16 + l15;
    const float2 b = wr_b[o];
#pragma unroll
    for (int v = 0; v < 8; v++) {
      const int m = v + 8 * lh;
      const int p = p0 + m;
      const int n = p >> 12, f = p & (FDIM - 1);
      const float2 s = spec[(size_t)p * HIDD + o];
      const float2 tv = tpi[(size_t)p * HIDD + o];
      float rev = cre[ot][v] + b.x + s.x + tv.x;
      float imv = cim[ot][v] + b.y + s.y + tv.y;
      euler_act(rev, imv, 5.0f);
      xco[(size_t)p * HIDD + o] = make_float2(rev, imv);
      xcT[((size_t)n * HIDD + o) * FDIM + f] = make_float2(rev, imv);
    }
  }
}

// ---------------------------------------------------------------------------
// Final MLP, fully fused per 16-point tile:
//   x1..x3 from mod GEMM (3 passes), h0 = act(i*pos.w0) analytic,
//   h = act(lin(h0+x1,w1)); h = act(lin(h+x2,w2)); out = Re(lin(h+x3,wend))
// ---------------------------------------------------------------------------
__global__ void k_final(const float2* __restrict__ xc,
                        const float2* __restrict__ mod_w, const float2* __restrict__ mod_b,
                        const float2* __restrict__ w0,
                        const float2* __restrict__ w1_w, const float2* __restrict__ w1_b,
                        const float2* __restrict__ w2_w, const float2* __restrict__ w2_b,
                        const float2* __restrict__ wend_w, const float2* __restrict__ wend_b,
                        float* __restrict__ outp) {
  __shared__ float lds[2][3][2][16][HIDD];   // [wave][xcL,xb,hb][re/im][m][o]
  const int lane = threadIdx.x & 31, wv = threadIdx.x >> 5;
  const int l15 = lane & 15, lh = lane >> 4;
  const int p0 = (blockIdx.x * 2 + wv) * 16;
  float* Xre = &lds[wv][0][0][0][0]; float* Xim = &lds[wv][0][1][0][0];
  float* Bre = &lds[wv][1][0][0][0]; float* Bim = &lds[wv][1][1][0][0];
  float* Hre = &lds[wv][2][0][0][0]; float* Him = &lds[wv][2][1][0][0];
  const float E1 = 0.3678794411714423f;
  const v8f vz = {0.f,0.f,0.f,0.f,0.f,0.f,0.f,0.f};
  v8f cre[4], cim[4];

  for (int e = lane; e < 16 * HIDD; e += 32) {
    const float2 v = xc[(size_t)p0 * HIDD + e];
    Xre[e] = v.x; Xim[e] = v.y;
  }
  __syncthreads();

  // x1 = mod pass 0 -> xb
  for (int i = 0; i < 4; i++) { cre[i] = vz; cim[i] = vz; }
  cgemm64(Xre, Xim, mod_w, 0, cre, cim, l15, lh);
#pragma unroll
  for (int ot = 0; ot < 4; ot++) {
    const int o = ot * 16 + l15;
    const float2 b = mod_b[o];
#pragma unroll
    for (int v = 0; v < 8; v++) {
      const int m = v + 8 * lh;
      Bre[m * HIDD + o] = cre[ot][v] + b.x;
      Bim[m * HIDD + o] = cim[ot][v] + b.y;
    }
  }
  __syncthreads();
  // xb += h0 = act(i * pos . w0_im)
  for (int e = lane; e < 16 * HIDD; e += 32) {
    const int m = e >> 6, dh = e & 63;
    const int p = p0 + m;
    const int f = p & (FDIM - 1);
    const float ft = (float)(f >> 8), fh = (float)((f >> 4) & 15), fw = (float)(f & 15);
    const float th = ft * w0[0 * HIDD + dh].y + fh * w0[1 * HIDD + dh].y + fw * w0[2 * HIDD + dh].y;
    float s, c; __sincosf(th, &s, &c);
    Bre[e] += E1 * c; Bim[e] += E1 * s;
  }
  __syncthreads();
  // hb = act(lin(xb, w1))
  for (int i = 0; i < 4; i++) { cre[i] = vz; cim[i] = vz; }
  cgemm64(Bre, Bim, w1_w, 0, cre, cim, l15, lh);
#pragma unroll
  for (int ot = 0; ot < 4; ot++) {
    const int o = ot * 16 + l15;
    const float2 b = w1_b[o];
#pragma unroll
    for (int v = 0; v < 8; v++) {
      const int m = v + 8 * lh;
      float rev = cre[ot][v] + b.x, imv = cim[ot][v] + b.y;
      euler_act(rev, imv, 1.0f);
      Hre[m * HIDD + o] = rev; Him[m * HIDD + o] = imv;
    }
  }
  __syncthreads();
  // hb += x2 (mod pass 1)
  for (int i = 0; i < 4; i++) { cre[i] = vz; cim[i] = vz; }
  cgemm64(Xre, Xim, mod_w, HIDD, cre, cim, l15, lh);
#pragma unroll
  for (int ot = 0; ot < 4; ot++) {
    const int o = ot * 16 + l15;
    const float2 b = mod_b[HIDD + o];
#pragma unroll
    for (int v = 0; v < 8; v++) {
      const int m = v + 8 * lh;
      Hre[m * HIDD + o] += cre[ot][v] + b.x;
      Him[m * HIDD + o] += cim[ot][v] + b.y;
    }
  }
  __syncthreads();
  // xb = act(lin(hb, w2))
  for (int i = 0; i < 4; i++) { cre[i] = vz; cim[i] = vz; }
  cgemm64(Hre, Him, w2_w, 0, cre, cim, l15, lh);
#pragma unroll
  for (int ot = 0; ot < 4; ot++) {
    const int o = ot * 16 + l15;
    const float2 b = w2_b[o];
#pragma unroll
    for (int v = 0; v < 8; v++) {
      const int m = v + 8 * lh;
      float rev = cre[ot][v] + b.x, imv = cim[ot][v] + b.y;
      euler_act(rev, imv, 1.0f);
      Bre[m * HIDD + o] = rev; Bim[m * HIDD + o] = imv;
    }
  }
  __syncthreads();
  // xb += x3 (mod pass 2)
  for (int i = 0; i < 4; i++) { cre[i] = vz; cim[i] = vz; }
  cgemm64(Xre, Xim, mod_w, 2 * HIDD, cre, cim, l15, lh);
#pragma unroll
  for (int ot = 0; ot < 4; ot++) {
    const int o = ot * 16 + l15;
    const float2 b = mod_b[2 * HIDD + o];
#pragma unroll
    for (int v = 0; v < 8; v++) {
      const int m = v + 8 * lh;
      Bre[m * HIDD + o] += cre[ot][v] + b.x;
      Bim[m * HIDD + o] += cim[ot][v] + b.y;
    }
  }
  __syncthreads();
  // wend (O=3, VALU), output Re only, layout [n][t][o][h*w]
  if (lh == 0) {
    const int m = l15;
    const int p = p0 + m;
    const int n = p >> 12, f = p & (FDIM - 1);
    const int tt = f >> 8, hw = f & 255;
    for (int o = 0; o < 3; o++) {
      float acc = wend_b[o].x;
      for (int i = 0; i < HIDD; i++) {
        const float2 w = wend_w[o * HIDD + i];
        acc += Bre[m * HIDD + i] * w.x - Bim[m * HIDD + i] * w.y;
      }
      outp[(((size_t)n * 16 + tt) * 3 + o) * 256 + hw] = acc;
    }
  }
}

// ---------------------------------------------------------------------------
extern "C" void kernel_launch(void* const* d_in, const int* in_sizes, int n_in,
                              void* d_out, int out_size, void* d_ws, size_t ws_size,
                              hipStream_t stream) {
  const float*  t      = (const float*) d_in[0];
  const float*  x      = (const float*) d_in[1];
  const float2* t_w    = (const float2*)d_in[2];
  const float2* tm0_w  = (const float2*)d_in[3];
  const float2* tm0_b  = (const float2*)d_in[4];
  const float2* tm1_w  = (const float2*)d_in[5];
  const float2* tm1_b  = (const float2*)d_in[6];
  const float2* proj_w = (const float2*)d_in[7];
  const float2* proj_b = (const float2*)d_in[8];
  const float2* rr[3]  = {(const float2*)d_in[9],  (const float2*)d_in[12], (const float2*)d_in[15]};
  const float2* wrw[3] = {(const float2*)d_in[10], (const float2*)d_in[13], (const float2*)d_in[16]};
  const float2* wrb[3] = {(const float2*)d_in[11], (const float2*)d_in[14], (const float2*)d_in[17]};
  const float2* mod_w  = (const float2*)d_in[18];
  const float2* mod_b  = (const float2*)d_in[19];
  const float2* w0     = (const float2*)d_in[20];
  const float2* w1_w   = (const float2*)d_in[21];
  const float2* w1_b   = (const float2*)d_in[22];
  const float2* w2_w   = (const float2*)d_in[23];
  const float2* w2_b   = (const float2*)d_in[24];
  const float2* wend_w = (const float2*)d_in[25];
  const float2* wend_b = (const float2*)d_in[26];

  float2* ws = (float2*)d_ws;
  const size_t C = (size_t)PTS * HIDD;   // 4,194,304 complex per buffer
  float2* tp   = ws;                     // 3*C  (tp1,tp2,tp3)
  float2* xcA  = ws + 3 * C;
  float2* xcB  = ws + 4 * C;
  float2* xcT  = ws + 5 * C;             // channel-major [n][D][f]
  float2* zf   = ws + 6 * C;             // [f][D][n]
  float2* zr   = ws + 7 * C;             // [n][k][f]
  float2* spec = ws + 8 * C;             // point-major [p][k]

  k_tp  <<<PTS / 32, 64, 0, stream>>>(t, t_w, tm0_w, tm0_b, tm1_w, tm1_b, tp);
  k_proj<<<PTS / 256, 256, 0, stream>>>(x, proj_w, proj_b, xcA, xcT);

  float2* cur = xcA; float2* nxt = xcB;
  for (int i = 0; i < 3; i++) {
    k_fft<0>   <<<NB * HIDD, 256, 0, stream>>>(xcT, zf);
    k_spectral <<<FDIM / 2, 64, 0, stream>>>(zf, rr[i], zr);
    k_fft<1>   <<<NB * HIDD, 256, 0, stream>>>(zr, spec);
    k_layeract <<<PTS / 32, 64, 0, stream>>>(cur, wrw[i], wrb[i], spec,
                                             tp + (size_t)i * C, nxt, xcT);
    float2* tmp = cur; cur = nxt; nxt = tmp;
  }
  k_final<<<PTS / 32, 64, 0, stream>>>(cur, mod_w, mod_b, w0, w1_w, w1_b,
                                       w2_w, w2_b, wend_w, wend_b, (float*)d_out);
  (void)in_sizes; (void)n_in; (void)out_size; (void)ws_size;
}